// GeometricEmbedding_36112085024915
// MI455X (gfx1250) — compile-verified
//
#include <hip/hip_runtime.h>
#include <math.h>

typedef float v2f __attribute__((ext_vector_type(2)));
typedef float v8f __attribute__((ext_vector_type(8)));

#define QN 100000
#define SN 100000

// ---------------------------------------------------------------- zero ws
__global__ void ge_zero_kernel(float* __restrict__ p, int n) {
    int i = blockIdx.x * blockDim.x + threadIdx.x;
    if (i < n) p[i] = 0.0f;
}

// ---------------------------------------------------------------- edge scatter
// acc[q*12]: {count, sum_d, sum_d2, sx, sy, sz, sxx, syy, szz, sxy, sxz, syz}
__global__ void ge_edge_kernel(const float* __restrict__ spos,
                               const float* __restrict__ qpos,
                               const int* __restrict__ qi,
                               const int* __restrict__ si,
                               float* __restrict__ acc, int E) {
    int e = blockIdx.x * blockDim.x + threadIdx.x;
    if (e >= E) return;
    int q = qi[e];
    int s = si[e];
    if ((unsigned)q >= (unsigned)QN || (unsigned)s >= (unsigned)SN) return;
    float qx = qpos[q * 3 + 0], qy = qpos[q * 3 + 1], qz = qpos[q * 3 + 2];
    float sx = spos[s * 3 + 0], sy = spos[s * 3 + 1], sz = spos[s * 3 + 2];
    float dx = sx - qx, dy = sy - qy, dz = sz - qz;
    float dist = sqrtf(dx * dx + dy * dy + dz * dz);
    float* a = acc + (size_t)q * 12;
    atomicAdd(a + 0, 1.0f);
    atomicAdd(a + 1, dist);
    atomicAdd(a + 2, dist * dist);
    atomicAdd(a + 3, sx);
    atomicAdd(a + 4, sy);
    atomicAdd(a + 5, sz);
    atomicAdd(a + 6, sx * sx);
    atomicAdd(a + 7, sy * sy);
    atomicAdd(a + 8, sz * sz);
    atomicAdd(a + 9, sx * sy);
    atomicAdd(a + 10, sx * sz);
    atomicAdd(a + 11, sy * sz);
}

// ---------------------------------------------------------------- 3x3 sym eig (descending)
__device__ inline void eig3_desc(float a00, float a11, float a22,
                                 float a01, float a02, float a12,
                                 float& e0, float& e1, float& e2) {
    float p1 = a01 * a01 + a02 * a02 + a12 * a12;
    float q = (a00 + a11 + a22) * (1.0f / 3.0f);
    if (p1 < 1e-30f) {  // diagonal: sort diag descending
        float x = a00, y = a11, z = a22, t;
        if (x < y) { t = x; x = y; y = t; }
        if (y < z) { t = y; y = z; z = t; }
        if (x < y) { t = x; x = y; y = t; }
        e0 = x; e1 = y; e2 = z;
        return;
    }
    float d0 = a00 - q, d1 = a11 - q, d2 = a22 - q;
    float p2 = d0 * d0 + d1 * d1 + d2 * d2 + 2.0f * p1;
    float p = sqrtf(p2 * (1.0f / 6.0f));
    float ip = 1.0f / p;
    float b00 = d0 * ip, b11 = d1 * ip, b22 = d2 * ip;
    float b01 = a01 * ip, b02 = a02 * ip, b12 = a12 * ip;
    float detB = b00 * (b11 * b22 - b12 * b12)
               - b01 * (b01 * b22 - b12 * b02)
               + b02 * (b01 * b12 - b11 * b02);
    float r = 0.5f * detB;
    r = fminf(1.0f, fmaxf(-1.0f, r));
    float phi = acosf(r) * (1.0f / 3.0f);
    float c0 = q + 2.0f * p * __cosf(phi);
    float c2 = q + 2.0f * p * __cosf(phi + 2.09439510239319549f);
    float c1 = 3.0f * q - c0 - c2;
    e0 = c0; e1 = c1; e2 = c2;
}

// ---------------------------------------------------------------- per-query feats + stats
__global__ void ge_feats_kernel(const float* __restrict__ qpos,
                                const float* __restrict__ acc,
                                float* __restrict__ feats,
                                double* __restrict__ dsum,
                                double* __restrict__ dsumsq, int Q) {
    __shared__ double sh[18];
    int tid = threadIdx.x;
    if (tid < 18) sh[tid] = 0.0;
    __syncthreads();
    int q = blockIdx.x * blockDim.x + tid;

    float f[9];
#pragma unroll
    for (int j = 0; j < 9; j++) f[j] = 0.0f;

    if (q < Q) {
        const float* a = acc + (size_t)q * 12;
        float cnt = a[0];
        if (cnt > 0.0f) {
            float inv = 1.0f / cnt;
            float Davg = a[1] * inv;
            float Dvar = fmaxf(a[2] * inv - Davg * Davg, 0.0f);
            float cx = a[3] * inv, cy = a[4] * inv, cz = a[5] * inv;
            float qx = qpos[q * 3 + 0], qy = qpos[q * 3 + 1], qz = qpos[q * 3 + 2];
            float cxx = a[6] * inv - cx * cx;
            float cyy = a[7] * inv - cy * cy;
            float czz = a[8] * inv - cz * cz;
            float cxy = a[9] * inv - cx * cy;
            float cxz = a[10] * inv - cx * cz;
            float cyz = a[11] * inv - cy * cz;
            float e0, e1, e2;
            eig3_desc(cxx, cyy, czz, cxy, cxz, cyz, e0, e1, e2);
            f[0] = cnt; f[1] = Davg; f[2] = Dvar;
            f[3] = cx - qx; f[4] = cy - qy; f[5] = cz - qz;
            f[6] = e0; f[7] = e1; f[8] = e2;
        }
        float* fr = feats + (size_t)q * 16;
#pragma unroll
        for (int j = 0; j < 9; j++) fr[j] = f[j];
#pragma unroll
        for (int j = 9; j < 16; j++) fr[j] = 0.0f;
    }

    // wave32 shuffle reduction in double
    double vs[9], vq[9];
#pragma unroll
    for (int j = 0; j < 9; j++) {
        vs[j] = (double)f[j];
        vq[j] = (double)f[j] * (double)f[j];
    }
#pragma unroll
    for (int off = 16; off > 0; off >>= 1) {
#pragma unroll
        for (int j = 0; j < 9; j++) {
            vs[j] += __shfl_down(vs[j], off, 32);
            vq[j] += __shfl_down(vq[j], off, 32);
        }
    }
    int lane = tid & 31;
    if (lane == 0) {
#pragma unroll
        for (int j = 0; j < 9; j++) {
            atomicAdd(&sh[j], vs[j]);
            atomicAdd(&sh[9 + j], vq[j]);
        }
    }
    __syncthreads();
    if (tid < 9) {
        atomicAdd(&dsum[tid], sh[tid]);
        atomicAdd(&dsumsq[tid], sh[9 + tid]);
    }
}

// ---------------------------------------------------------------- mean / inv-std (padded to 16)
// j in [9,16): mean = 0, invstd = 0 so padded columns normalize to exactly 0.
__global__ void ge_finalize_kernel(const double* __restrict__ dsum,
                                   const double* __restrict__ dsumsq,
                                   float* __restrict__ meanp,
                                   float* __restrict__ invp, int Q) {
    int j = threadIdx.x;
    if (j >= 16) return;
    if (j < 9) {
        double m = dsum[j] / (double)Q;
        double var = (dsumsq[j] - (double)Q * m * m) / (double)(Q - 1);
        if (var < 0.0) var = 0.0;
        double sd = sqrt(var);
        if (sd < 1e-6) sd = 1.0;
        meanp[j] = (float)m;
        invp[j] = (float)(1.0 / sd);
    } else {
        meanp[j] = 0.0f;
        invp[j] = 0.0f;
    }
}

// ---------------------------------------------------------------- fused normalize + MLP (fp32 WMMA)
// Each wave: 16 query rows. GEMM1: 16x12(K pad 9)x64, GEMM2: 16x64x128.
__global__ __launch_bounds__(128) void ge_mlp_wmma_kernel(
    const float* __restrict__ feats, const float* __restrict__ meanp,
    const float* __restrict__ invp, const float* __restrict__ W1,
    const float* __restrict__ b1, const float* __restrict__ W2,
    const float* __restrict__ b2, float* __restrict__ out, int Q) {
    __shared__ float sW2t[64 * 128];        // W2 transposed: [k][n]
    __shared__ float sW1t[16 * 64];         // W1 transposed, zero-padded: [k][n]
    __shared__ float sHA[4][16 * 68];       // per-wave hidden tile, padded stride 68

    int tid = threadIdx.x;
    // stage W2^T cooperatively (coalesced reads of W2[128][64])
    for (int idx = tid; idx < 64 * 128; idx += 128) {
        int n = idx >> 6;
        int k = idx & 63;
        sW2t[k * 128 + n] = W2[idx];
    }
    // stage zero-padded W1^T: sW1t[k][n] = k<9 ? W1[n][k] : 0
    for (int idx = tid; idx < 16 * 64; idx += 128) {
        int k = idx >> 6;
        int n = idx & 63;
        sW1t[idx] = (k < 9) ? W1[n * 9 + k] : 0.0f;
    }
    __syncthreads();

    int wave = tid >> 5;
    int lane = tid & 31;
    int tile = blockIdx.x * 4 + wave;
    if (tile * 16 >= Q) return;             // whole-wave guard (Q % 16 == 0)

    int lrow = lane & 15;
    bool hi = lane >= 16;
    int koff = hi ? 2 : 0;
    int row = tile * 16 + lrow;
    const float* frow = feats + (size_t)row * 16;

    // ---- GEMM1: h = relu(featsN @ W1^T + b1) ----
    v8f c1[4];
#pragma unroll
    for (int nt = 0; nt < 4; nt++)
#pragma unroll
        for (int r = 0; r < 8; r++) c1[nt][r] = 0.0f;

#pragma unroll
    for (int kc = 0; kc < 3; kc++) {        // K = 12 covers stat_dim 9 (rest zero)
        int k0 = kc * 4 + koff;
        v2f a;
        // padded mean/invstd: for k>=9 invp==0 and frow==0 -> exact 0, no branch
        a.x = (frow[k0] - meanp[k0]) * invp[k0];
        a.y = (frow[k0 + 1] - meanp[k0 + 1]) * invp[k0 + 1];
#pragma unroll
        for (int nt = 0; nt < 4; nt++) {
            int n = nt * 16 + lrow;
            v2f b;
            b.x = sW1t[k0 * 64 + n];
            b.y = sW1t[(k0 + 1) * 64 + n];
            c1[nt] = __builtin_amdgcn_wmma_f32_16x16x4_f32(
                false, a, false, b, (short)0, c1[nt], false, false);
        }
    }

    // bias + relu, stage hidden tile to LDS (C-layout -> memory layout)
    float* ha = &sHA[wave][0];
#pragma unroll
    for (int nt = 0; nt < 4; nt++) {
        int n = nt * 16 + lrow;
        float bb = b1[n];
#pragma unroll
        for (int r = 0; r < 8; r++) {
            int rr = r + (hi ? 8 : 0);
            ha[rr * 68 + n] = fmaxf(c1[nt][r] + bb, 0.0f);
        }
    }

    // ---- GEMM2: out = h @ W2^T + b2 ----
    v8f c2[8];
#pragma unroll
    for (int nt = 0; nt < 8; nt++) {
        float bv = b2[nt * 16 + lrow];
#pragma unroll
        for (int r = 0; r < 8; r++) c2[nt][r] = bv;
    }
#pragma unroll
    for (int kc = 0; kc < 16; kc++) {
        int k0 = kc * 4 + koff;
        v2f a;
        a.x = ha[lrow * 68 + k0];
        a.y = ha[lrow * 68 + k0 + 1];
#pragma unroll
        for (int nt = 0; nt < 8; nt++) {
            int n = nt * 16 + lrow;
            v2f b;
            b.x = sW2t[k0 * 128 + n];
            b.y = sW2t[(k0 + 1) * 128 + n];
            c2[nt] = __builtin_amdgcn_wmma_f32_16x16x4_f32(
                false, a, false, b, (short)0, c2[nt], false, false);
        }
    }

    // store 16x128 output tile
    float* orow = out + (size_t)tile * 16 * 128;
#pragma unroll
    for (int nt = 0; nt < 8; nt++) {
#pragma unroll
        for (int r = 0; r < 8; r++) {
            int m = r + (hi ? 8 : 0);
            orow[m * 128 + nt * 16 + lrow] = c2[nt][r];
        }
    }
}

// ---------------------------------------------------------------- host
extern "C" void kernel_launch(void* const* d_in, const int* in_sizes, int n_in,
                              void* d_out, int out_size, void* d_ws, size_t ws_size,
                              hipStream_t stream) {
    const float* spos = (const float*)d_in[0];
    const float* qpos = (const float*)d_in[1];
    const int* ei = (const int*)d_in[2];
    const float* W1 = (const float*)d_in[3];
    const float* b1 = (const float*)d_in[4];
    const float* W2 = (const float*)d_in[5];
    const float* b2 = (const float*)d_in[6];
    float* out = (float*)d_out;

    const int Q = QN;
    const int E = in_sizes[2] / 2;

    char* ws = (char*)d_ws;
    double* dsum = (double*)ws;                      // 9 doubles @ 0
    double* dsumsq = dsum + 9;                       // 9 doubles @ 72
    float* acc = (float*)(ws + 256);                 // Q*12 floats
    float* feats = acc + (size_t)Q * 12;             // Q*16 floats
    float* meanp = feats + (size_t)Q * 16;           // 16 floats (padded)
    float* invp = meanp + 16;                        // 16 floats (padded)

    // zero header (64 floats = 256B) + accumulators
    int nz = 64 + Q * 12;
    ge_zero_kernel<<<(nz + 255) / 256, 256, 0, stream>>>((float*)ws, nz);

    ge_edge_kernel<<<(E + 255) / 256, 256, 0, stream>>>(spos, qpos, ei, ei + E, acc, E);

    ge_feats_kernel<<<(Q + 255) / 256, 256, 0, stream>>>(qpos, acc, feats, dsum, dsumsq, Q);

    ge_finalize_kernel<<<1, 32, 0, stream>>>(dsum, dsumsq, meanp, invp, Q);

    int tiles = (Q + 15) / 16;                       // 6250
    ge_mlp_wmma_kernel<<<(tiles + 3) / 4, 128, 0, stream>>>(
        feats, meanp, invp, W1, b1, W2, b2, out, Q);
}